// QuestionPassageCoAttention_58488864637419
// MI455X (gfx1250) — compile-verified
//
#include <hip/hip_runtime.h>
#include <hip/hip_bf16.h>
#include <math.h>

// ---------------------------------------------------------------------------
// QuestionPassageCoAttention for MI455X (gfx1250).
// Batch dim is embarrassingly parallel across the 500-step recurrence, so each
// workgroup owns (direction, 16-batch tile) and runs the whole scan with LDS
// state + workgroup barriers. GEMMs use V_WMMA_F32_16X16X4_F32 (f32 in/out: no
// precision loss across 500 steps; chip is chain-latency bound, not peak
// bound). Weights repacked once to [kblk][n][4] so each lane's B-fragment is a
// coalesced 8-byte load (L2-resident, ~1MB). h-independent terms (Qbase, PU)
// precomputed chip-wide with WMMA. Per-step PU tile is DMA'd into LDS with the
// Tensor Data Mover (tensor_load_to_lds + s_wait_tensorcnt), overlapped with
// the pv WMMA GEMM.
// ---------------------------------------------------------------------------

#define H_   75
#define B_   256
#define LQ_  64
#define LP_  500
#define D2   150   // 2H
#define D3   225   // 3H
#define D4   300   // 4H
#define HP   80    // H padded to 5 wmma n-tiles
#define D3P  240   // 3H padded
#define D4P  304   // 4H padded
#define QSTR 76    // Qbase row stride (76 floats = 304B, 16B multiple)

typedef float v2f __attribute__((ext_vector_type(2)));
typedef float v8f __attribute__((ext_vector_type(8)));
typedef unsigned int v4u __attribute__((ext_vector_type(4)));
typedef int v4i __attribute__((ext_vector_type(4)));
typedef int v8i __attribute__((ext_vector_type(8)));

// ---- WMMA helpers (f32 16x16x4) -------------------------------------------

__device__ __forceinline__ v8f wmma4(v2f a, v2f b, v8f c) {
  // (neg_a, A, neg_b, B, c_mod, C, reuse_a, reuse_b)
  return __builtin_amdgcn_wmma_f32_16x16x4_f32(false, a, false, b, (short)0, c,
                                               false, false);
}

// A (16 x K) in LDS, row stride lda (even). B packed in global as
// B4[(kblk*Npack + n)*4 + kin] == B[kblk*4+kin][n].
__device__ __forceinline__ v8f gemm_tile(const float* __restrict__ As, int lda,
                                         const float* __restrict__ B4, int Npack,
                                         int ntile, int kblks) {
  const int lane = threadIdx.x & 31;
  const int m    = lane & 15;
  const int koff = (lane >> 4) * 2;
  const int n    = ntile * 16 + m;
  v8f acc = {};
  const float* a0 = As + m * lda + koff;
  const float* b0 = B4 + (size_t)n * 4 + koff;
  for (int kb = 0; kb < kblks; ++kb) {
    v2f a = *(const v2f*)(a0 + kb * 4);
    v2f b = *(const v2f*)(b0 + (size_t)kb * Npack * 4);
    acc = wmma4(a, b, acc);
  }
  return acc;
}

// D layout: VGPR r -> M = r + 8*(lane>=16), N = ntile*16 + (lane&15)
__device__ __forceinline__ void store_tile(float* __restrict__ D, int ldd,
                                           int ntile, v8f acc) {
  const int lane  = threadIdx.x & 31;
  const int n     = ntile * 16 + (lane & 15);
  const int mbase = (lane >> 4) * 8;
#pragma unroll
  for (int r = 0; r < 8; ++r) D[(mbase + r) * ldd + n] = acc[r];
}

// ---- TDM: 1-D contiguous global -> LDS copy (ndwords DWORDs) --------------

__device__ __forceinline__ void tdm_load_1d(unsigned lds_addr,
                                            unsigned long long gaddr,
                                            int ndwords) {
#if __has_builtin(__builtin_amdgcn_tensor_load_to_lds)
  // D# group0: count=1, lds_addr, 57-bit global addr, type=2 ("image")
  v4u g0;
  g0.x = 1u;
  g0.y = lds_addr;
  g0.z = (unsigned)(gaddr & 0xffffffffull);
  g0.w = (unsigned)((gaddr >> 32) & 0x01ffffffull) | (2u << 30);
  // D# group1: data_size=4B; tensor_dim0=tile_dim0=stride=ndwords; 1-D tile
  unsigned nd = (unsigned)ndwords;
  v8i g1;
  g1[0] = (int)(2u << 16);                       // data_size = 2 (4 bytes)
  g1[1] = (int)((nd & 0xffffu) << 16);           // tensor_dim0[15:0]
  g1[2] = (int)((nd >> 16) | (1u << 16));        // tensor_dim0[31:16] | tensor_dim1=1
  g1[3] = (int)((nd & 0xffffu) << 16);           // tile_dim0
  g1[4] = 0;                                     // tile_dim1/2 unused
  g1[5] = (int)nd;                               // tensor_dim0_stride[31:0]
  g1[6] = 0;
  g1[7] = 0;
  v4i z4 = {0, 0, 0, 0};
#if __clang_major__ >= 23
  v8i z8 = {0, 0, 0, 0, 0, 0, 0, 0};
  __builtin_amdgcn_tensor_load_to_lds(g0, g1, z4, z4, z8, 0);
#else
  __builtin_amdgcn_tensor_load_to_lds(g0, g1, z4, z4, 0);
#endif
#endif
}

// ---- fast activations ------------------------------------------------------

__device__ __forceinline__ float fast_tanh(float x) {
  x = fminf(fmaxf(x, -12.f), 12.f);
  float e = __expf(2.f * x);
  return (e - 1.f) / (e + 1.f);
}
__device__ __forceinline__ float fast_sigmoid(float x) {
  x = fminf(fmaxf(x, -30.f), 30.f);
  return 1.f / (1.f + __expf(-x));
}

// ---- packing kernels -------------------------------------------------------

__global__ void pack_w4_kernel(const float* __restrict__ W,
                               float* __restrict__ out4,
                               int N, int K, int Npack, int Kblks) {
  int total = Kblks * Npack * 4;
  for (int idx = blockIdx.x * blockDim.x + threadIdx.x; idx < total;
       idx += gridDim.x * blockDim.x) {
    int ki = idx & 3;
    int n  = (idx >> 2) % Npack;
    int kb = (idx >> 2) / Npack;
    int k  = kb * 4 + ki;
    out4[idx] = (n < N && k < K) ? W[(size_t)n * K + k] : 0.f;
  }
}

// QT[b][d][q] = quesEnc[q][b][d]
__global__ void pack_qt_kernel(const float* __restrict__ quesEnc,
                               float* __restrict__ QT) {
  int total = B_ * D2 * LQ_;
  for (int idx = blockIdx.x * blockDim.x + threadIdx.x; idx < total;
       idx += gridDim.x * blockDim.x) {
    int q = idx & 63;
    int d = (idx >> 6) % D2;
    int b = (idx >> 6) / D2;
    QT[idx] = quesEnc[((size_t)q * B_ + b) * D2 + d];
  }
}

// ---- precompute GEMM: 16 rows/WG, N=80 (5 tiles), K=152 (38 blks) ----------
__global__ __launch_bounds__(256, 2)
void gemm_rows16_kernel(const float* __restrict__ A0, int mode,
                        const float* __restrict__ B4,
                        float* __restrict__ out, int ostride,
                        const float* __restrict__ bias0,
                        const float* __restrict__ bias1,
                        const float* __restrict__ bias2,
                        int addBias, int kblks) {
  __shared__ __align__(16) float As[16 * 152];
  const int tid = threadIdx.x;
  const int row0 = blockIdx.x * 16;
  for (int i = tid; i < 16 * 152; i += 256) {
    int r = i / 152, k = i % 152;
    float v = 0.f;
    if (k < D2) {
      int m = row0 + r;
      const float* arow = (mode == 0)
          ? (A0 + (size_t)m * D2)
          : (A0 + ((size_t)(m & 63) * B_ + (m >> 6)) * D2);
      v = arow[k];
    }
    As[i] = v;
  }
  __syncthreads();
  const int wv = tid >> 5;
  if (wv < 5) {
    v8f acc = gemm_tile(As, 152, B4, HP, wv, kblks);
    const int lane  = tid & 31;
    const int n     = wv * 16 + (lane & 15);
    const int mbase = (lane >> 4) * 8;
    if (n < ostride) {
      float bb = (addBias && n < H_) ? (bias0[n] + bias1[n] + bias2[n]) : 0.f;
#pragma unroll
      for (int r = 0; r < 8; ++r)
        out[(size_t)(row0 + mbase + r) * ostride + n] = acc[r] + bb;
    }
  }
}

// ---- persistent scan kernel ------------------------------------------------
// grid = 32: blockIdx>>4 = direction, blockIdx&15 = batch tile (16 rows).

__global__ __launch_bounds__(512, 1)
void scan_kernel(const float* __restrict__ passEnc,
                 const float* __restrict__ Qbase,
                 const float* __restrict__ PU,
                 const float* __restrict__ QT,
                 const float* __restrict__ Vt,
                 const float* __restrict__ WPvT4,
                 const float* __restrict__ WgT4,
                 const float* __restrict__ WihT4f,
                 const float* __restrict__ WihT4r,
                 const float* __restrict__ WhhT4f,
                 const float* __restrict__ WhhT4r,
                 const float* __restrict__ Wg_b,
                 const float* __restrict__ bih_f,
                 const float* __restrict__ bhh_f,
                 const float* __restrict__ bih_r,
                 const float* __restrict__ bhh_r,
                 const float* __restrict__ h0_f,
                 const float* __restrict__ h0_r,
                 float* __restrict__ out) {
  const int tid  = threadIdx.x;
  const int wv   = tid >> 5;
  const int lane = tid & 31;
  const int dir  = blockIdx.x >> 4;
  const int bg0  = (blockIdx.x & 15) * 16;

  const float* WihT4 = dir ? WihT4r : WihT4f;
  const float* WhhT4 = dir ? WhhT4r : WhhT4f;
  const float* bih   = dir ? bih_r  : bih_f;
  const float* bhh   = dir ? bhh_r  : bhh_f;
  const float* h0    = dir ? h0_r   : h0_f;

  __shared__ __align__(16) float s_cc[16 * D4P];  // [p | c | pad4] then x
  __shared__ __align__(16) float s_pool[7680];    // aliased pool
  __shared__ __align__(16) float s_h[16 * HP];
  __shared__ __align__(16) float s_v[16 * HP];
  __shared__ __align__(16) float s_bWg[D4P];
  __shared__ __align__(16) float s_bih[D3P];
  __shared__ __align__(16) float s_bhh[D3P];

  float* s_gpre = s_pool;             // 16*304  (P5..P6)
  float* s_gi   = s_pool;             // 16*240  (P7..P8)
  float* s_gh   = s_pool + 16 * D3P;  // 16*240  (P7..P8)
  float* s_s    = s_pool;             // 16*64   (P2..P4)
  float* s_upv  = s_pool + 1024;      // 16*80   (P1..P2)
  float* s_pu   = s_pool + 2304;      // 16*80   (P1..P2, TDM dest)

  for (int i = tid; i < 16 * HP; i += 512) {
    int b = i / HP, j = i % HP;
    s_h[i] = (j < H_) ? h0[(size_t)(bg0 + b) * H_ + j] : 0.f;
    s_v[i] = (j < H_) ? Vt[(size_t)(bg0 + b) * H_ + j] : 0.f;
  }
  for (int i = tid; i < D4P; i += 512) s_bWg[i] = (i < D4) ? Wg_b[i] : 0.f;
  for (int i = tid; i < D3P; i += 512) {
    s_bih[i] = (i < D3) ? bih[i] : 0.f;
    s_bhh[i] = (i < D3) ? bhh[i] : 0.f;
  }
  __syncthreads();

  for (int t = 0; t < LP_; ++t) {
    const int ta = dir ? (LP_ - 1 - t) : t;

    // ---- P1: TDM PU tile -> LDS; p -> cc[:, :150]; upv = h @ WPv^T --------
#if __has_builtin(__builtin_amdgcn_tensor_load_to_lds)
    if (wv == 0) {
      tdm_load_1d((unsigned)(size_t)(void*)s_pu,
                  (unsigned long long)(size_t)(const void*)
                      &PU[((size_t)ta * B_ + bg0) * HP],
                  16 * HP);
    }
#else
    for (int i = tid; i < 16 * HP; i += 512)
      s_pu[i] = PU[((size_t)ta * B_ + bg0) * HP + i];
#endif
    for (int i = tid; i < 16 * (D2 / 2); i += 512) {  // float2 copy of p
      int b = i / (D2 / 2), k2 = i % (D2 / 2);
      ((float2*)(s_cc + b * D4P))[k2] =
          ((const float2*)(passEnc + ((size_t)ta * B_ + bg0 + b) * D2))[k2];
    }
    if (tid < 64) s_cc[(tid >> 2) * D4P + D4 + (tid & 3)] = 0.f;
    if (wv < 5) {
      v8f acc = gemm_tile(s_h, HP, WPvT4, HP, wv, 19);
      store_tile(s_upv, HP, wv, acc);
    }
    if (t + 1 < LP_ && tid < 16) {
      int tn = dir ? (LP_ - 2 - t) : (t + 1);
      __builtin_prefetch(&passEnc[((size_t)tn * B_ + bg0 + tid) * D2], 0, 0);
    }
#if __has_builtin(__builtin_amdgcn_s_wait_tensorcnt)
    if (wv == 0) __builtin_amdgcn_s_wait_tensorcnt(0);
#endif
    __syncthreads();

    // ---- P2: s[b][q] = sum_h tanh(Qbase + pu + pv) * v  (float4, no tail:
    //      column 75 of every operand is zero by construction) -------------
    for (int p = tid; p < 16 * LQ_; p += 512) {
      int b = p >> 6, q = p & 63;
      const float4* qb =
          (const float4*)(Qbase + ((size_t)(bg0 + b) * LQ_ + q) * QSTR);
      const float4* uv = (const float4*)(s_upv + b * HP);
      const float4* pu = (const float4*)(s_pu + b * HP);
      const float4* vv = (const float4*)(s_v + b * HP);
      float acc = 0.f;
      for (int h4 = 0; h4 < QSTR / 4; ++h4) {
        float4 a = qb[h4], u = uv[h4], w = pu[h4], vx = vv[h4];
        acc += fast_tanh(a.x + u.x + w.x) * vx.x;
        acc += fast_tanh(a.y + u.y + w.y) * vx.y;
        acc += fast_tanh(a.z + u.z + w.z) * vx.z;
        acc += fast_tanh(a.w + u.w + w.w) * vx.w;
      }
      s_s[p] = acc;
    }
    __syncthreads();

    // ---- P3: softmax over q (one thread per batch row) --------------------
    if (tid < 16) {
      float* sr = s_s + tid * LQ_;
      float mx = sr[0];
      for (int q = 1; q < LQ_; ++q) mx = fmaxf(mx, sr[q]);
      float sum = 0.f;
      for (int q = 0; q < LQ_; ++q) { float e = __expf(sr[q] - mx); sr[q] = e; sum += e; }
      float inv = 1.f / sum;
      for (int q = 0; q < LQ_; ++q) sr[q] *= inv;
    }
    __syncthreads();

    // ---- P4: context c[b][d] -> cc[:, 150:300]  (float4 over q) -----------
    for (int p = tid; p < 16 * D2; p += 512) {
      int b = p / D2, d = p % D2;
      const float4* qt =
          (const float4*)(QT + ((size_t)(bg0 + b) * D2 + d) * LQ_);
      const float4* ar = (const float4*)(s_s + b * LQ_);
      float acc = 0.f;
      for (int q4 = 0; q4 < LQ_ / 4; ++q4) {
        float4 a = ar[q4], x = qt[q4];
        acc += a.x * x.x + a.y * x.y + a.z * x.z + a.w * x.w;
      }
      s_cc[b * D4P + D2 + d] = acc;
    }
    __syncthreads();

    // ---- P5: gpre = cc @ Wg^T  (19 n-tiles, K=304) ------------------------
    for (int nt = wv; nt < 19; nt += 16) {
      v8f acc = gemm_tile(s_cc, D4P, WgT4, D4P, nt, 76);
      store_tile(s_gpre, D4P, nt, acc);
    }
    __syncthreads();

    // ---- P6: x = sigmoid(gpre + Wg_b) * cc (in place) ---------------------
    for (int i = tid; i < 16 * D4; i += 512) {
      int b = i / D4, n = i % D4;
      float g = fast_sigmoid(s_gpre[b * D4P + n] + s_bWg[n]);
      s_cc[b * D4P + n] *= g;
    }
    __syncthreads();

    // ---- P7: gi = x @ Wih^T (15 tiles, K=304); gh = h @ Whh^T (K=76) ------
    if (wv < 15) {
      v8f acc = gemm_tile(s_cc, D4P, WihT4, D3P, wv, 76);
      store_tile(s_gi, D3P, wv, acc);
    }
    if (wv >= 1) {
      v8f acc = gemm_tile(s_h, HP, WhhT4, D3P, wv - 1, 19);
      store_tile(s_gh, D3P, wv - 1, acc);
    }
    __syncthreads();

    // ---- P8: GRU cell + output -------------------------------------------
    for (int i = tid; i < 16 * H_; i += 512) {
      int b = i / H_, j = i % H_;
      const float* gib = s_gi + b * D3P;
      const float* ghb = s_gh + b * D3P;
      float r = fast_sigmoid(gib[j] + s_bih[j] + ghb[j] + s_bhh[j]);
      float z = fast_sigmoid(gib[H_ + j] + s_bih[H_ + j] +
                             ghb[H_ + j] + s_bhh[H_ + j]);
      float nn = fast_tanh(gib[2 * H_ + j] + s_bih[2 * H_ + j] +
                           r * (ghb[2 * H_ + j] + s_bhh[2 * H_ + j]));
      float hn = (1.f - z) * nn + z * s_h[b * HP + j];
      s_h[b * HP + j] = hn;
      out[((size_t)t * B_ + bg0 + b) * (2 * H_) + dir * H_ + j] = hn;
    }
    __syncthreads();
  }
}

// ---- host launcher ---------------------------------------------------------

extern "C" void kernel_launch(void* const* d_in, const int* in_sizes, int n_in,
                              void* d_out, int out_size, void* d_ws,
                              size_t ws_size, hipStream_t stream) {
  const float* quesEnc = (const float*)d_in[0];
  const float* passEnc = (const float*)d_in[1];
  const float* WQu_w   = (const float*)d_in[2];
  const float* WQu_b   = (const float*)d_in[3];
  const float* WPu_w   = (const float*)d_in[4];
  const float* WPu_b   = (const float*)d_in[5];
  const float* WPv_w   = (const float*)d_in[6];
  const float* WPv_b   = (const float*)d_in[7];
  const float* Wg_w    = (const float*)d_in[8];
  const float* Wg_b    = (const float*)d_in[9];
  const float* Vt      = (const float*)d_in[10];
  const float* Wih_f   = (const float*)d_in[11];
  const float* Whh_f   = (const float*)d_in[12];
  const float* bih_f   = (const float*)d_in[13];
  const float* bhh_f   = (const float*)d_in[14];
  const float* Wih_r   = (const float*)d_in[15];
  const float* Whh_r   = (const float*)d_in[16];
  const float* bih_r   = (const float*)d_in[17];
  const float* bhh_r   = (const float*)d_in[18];
  const float* h0_f    = (const float*)d_in[19];
  const float* h0_r    = (const float*)d_in[20];
  float* out = (float*)d_out;
  float* ws  = (float*)d_ws;

  // workspace layout (floats)
  size_t o = 0;
  float* QBASE = ws + o;  o += (size_t)LQ_ * B_ * QSTR;   // 1,245,184
  float* PUb   = ws + o;  o += (size_t)LP_ * B_ * HP;     // 10,240,000
  float* QT    = ws + o;  o += (size_t)B_ * D2 * LQ_;     // 2,457,600
  float* WQUT4 = ws + o;  o += (size_t)38 * HP * 4;
  float* WPUT4 = ws + o;  o += (size_t)38 * HP * 4;
  float* WPVT4 = ws + o;  o += (size_t)19 * HP * 4;
  float* WGT4  = ws + o;  o += (size_t)76 * D4P * 4;
  float* WIHT4F = ws + o; o += (size_t)76 * D3P * 4;
  float* WIHT4R = ws + o; o += (size_t)76 * D3P * 4;
  float* WHHT4F = ws + o; o += (size_t)19 * D3P * 4;
  float* WHHT4R = ws + o; o += (size_t)19 * D3P * 4;
  (void)ws_size; (void)in_sizes; (void)n_in; (void)out_size;

  const int TB = 256;
  auto blocks = [&](size_t total) { return (int)((total + TB - 1) / TB); };

  pack_qt_kernel<<<blocks((size_t)B_ * D2 * LQ_), TB, 0, stream>>>(quesEnc, QT);

  pack_w4_kernel<<<blocks((size_t)38 * HP * 4), TB, 0, stream>>>(WQu_w, WQUT4, H_, D2, HP, 38);
  pack_w4_kernel<<<blocks((size_t)38 * HP * 4), TB, 0, stream>>>(WPu_w, WPUT4, H_, D2, HP, 38);
  pack_w4_kernel<<<blocks((size_t)19 * HP * 4), TB, 0, stream>>>(WPv_w, WPVT4, H_, H_, HP, 19);
  pack_w4_kernel<<<blocks((size_t)76 * D4P * 4), TB, 0, stream>>>(Wg_w, WGT4, D4, D4, D4P, 76);
  pack_w4_kernel<<<blocks((size_t)76 * D3P * 4), TB, 0, stream>>>(Wih_f, WIHT4F, D3, D4, D3P, 76);
  pack_w4_kernel<<<blocks((size_t)76 * D3P * 4), TB, 0, stream>>>(Wih_r, WIHT4R, D3, D4, D3P, 76);
  pack_w4_kernel<<<blocks((size_t)19 * D3P * 4), TB, 0, stream>>>(Whh_f, WHHT4F, D3, H_, D3P, 19);
  pack_w4_kernel<<<blocks((size_t)19 * D3P * 4), TB, 0, stream>>>(Whh_r, WHHT4R, D3, H_, D3P, 19);

  // Qbase[b][q][:] = quesEnc[q][b] @ WQu^T + (WQu_b + WPu_b + WPv_b)
  gemm_rows16_kernel<<<(LQ_ * B_) / 16, 256, 0, stream>>>(
      quesEnc, 1, WQUT4, QBASE, QSTR, WQu_b, WPu_b, WPv_b, 1, 38);
  // PU[t][b][:] = passEnc[t][b] @ WPu^T
  gemm_rows16_kernel<<<(LP_ * B_) / 16, 256, 0, stream>>>(
      passEnc, 0, WPUT4, PUb, HP, WQu_b, WQu_b, WQu_b, 0, 38);

  // persistent bidirectional scan: 2 dirs x 16 batch tiles
  scan_kernel<<<32, 512, 0, stream>>>(
      passEnc, QBASE, PUb, QT, Vt, WPVT4, WGT4, WIHT4F, WIHT4R, WHHT4F,
      WHHT4R, Wg_b, bih_f, bhh_f, bih_r, bhh_r, h0_f, h0_r, out);
}